// Decoder_26276609917594
// MI455X (gfx1250) — compile-verified
//
#include <hip/hip_runtime.h>
#include <hip/hip_bf16.h>

#define D_ACTS   768
#define D_FEAT   6144
#define N_LAYERS 12
#define BATCH    32
#define LSTR     (N_LAYERS * D_FEAT)   // 73728, feature stride per batch row

typedef __attribute__((ext_vector_type(2))) float v2f;
typedef __attribute__((ext_vector_type(8))) float v8f;

struct WPtrs { const float* w[N_LAYERS]; };

// ---------------------------------------------------------------------------
// Kernel 1: recons[b,l,a] = sum_{k<=l} sum_d F[b,k,d] * W_l[k,d,a]
// Grid: 12 layers (heavy first) x 24 tiles of 32 cols. Block: 512 thr = 16 waves.
// Wave (q,h): q = K-chunk (0..7, uniform via readfirstlane -> scalar loop),
// h = 16-col half. Each wave: 32x16 tile via 2x V_WMMA_F32_16X16X4_F32 per
// K=4 step (B frag reused across both M halves -> 16 FLOP/B on HBM stream).
// ---------------------------------------------------------------------------
__global__ __launch_bounds__(512) void decoder_gemm(
    const float* __restrict__ feat, WPtrs wp, float* __restrict__ out)
{
    __shared__ float tile[8][32][32];            // [kchunk][row(b)][col] 32 KB

    const int bid  = blockIdx.x;
    const int l    = (N_LAYERS - 1) - (bid / 24); // heavy layers first
    const int n0   = (bid % 24) * 32;
    const int tid  = threadIdx.x;
    // wave id is uniform within a wave: force it into an SGPR so the K-loop
    // bounds are scalar (s_cbranch loop, unrollable) instead of exec-masked.
    const int wave = __builtin_amdgcn_readfirstlane(tid >> 5);
    const int q    = wave >> 1;                  // K-chunk 0..7 (SGPR)
    const int h    = wave & 1;                   // N half 0..1  (SGPR)
    const int lane   = tid & 31;
    const int lane15 = lane & 15;
    const int ksel   = (lane >> 4) << 1;         // 0 or 2 (K sub-select in frag)
    const int col    = n0 + h * 16 + lane15;

    const float* __restrict__ W = wp.w[l];

    const int Kwave  = (l + 1) * (D_FEAT / 8);   // K elements per wave
    const int kstart = q * Kwave;
    const int kend   = kstart + Kwave;

    v8f c_lo = {0.f,0.f,0.f,0.f,0.f,0.f,0.f,0.f};
    v8f c_hi = {0.f,0.f,0.f,0.f,0.f,0.f,0.f,0.f};

    const float* fA = feat + lane15 * LSTR;      // M = lane15 (+16 for hi frag)

    const int kl0 = kstart / D_FEAT;
    const int klN = (kend - 1) / D_FEAT;
    for (int kl = kl0; kl <= klN; ++kl) {
        int dlo = kstart - kl * D_FEAT; if (dlo < 0) dlo = 0;
        int dhi = kend   - kl * D_FEAT; if (dhi > D_FEAT) dhi = D_FEAT;
        const float* a0 = fA + kl * D_FEAT + ksel;
        const float* b0 = W + (size_t)(kl * D_FEAT) * D_ACTS + col;
        #pragma unroll 4
        for (int d = dlo; d < dhi; d += 4) {
            // A frags: 16x4 f32, per-lane pair of consecutive d -> b64 load
            v2f a_lo = *(const v2f*)(a0 + d);
            v2f a_hi = *(const v2f*)(a0 + d + 16 * LSTR);
            // B frag: 4x16 f32, rows d+ksel and d+ksel+1, col fixed per lane
            const float* bp = b0 + (size_t)(d + ksel) * D_ACTS;
            v2f b;
            b.x = bp[0];
            b.y = bp[D_ACTS];
            c_lo = __builtin_amdgcn_wmma_f32_16x16x4_f32(
                false, a_lo, false, b, (short)0, c_lo, false, false);
            c_hi = __builtin_amdgcn_wmma_f32_16x16x4_f32(
                false, a_hi, false, b, (short)0, c_hi, false, false);
        }
    }

    // Scatter fragments to LDS in logical (row, col) layout.
    // C/D layout: VGPR v -> M=v (lanes 0-15), M=v+8 (lanes 16-31); N = lane&15.
    const int rbase = (lane >= 16) ? 8 : 0;
    const int ccol  = h * 16 + lane15;
    #pragma unroll
    for (int v = 0; v < 8; ++v) {
        tile[q][rbase + v][ccol]      = c_lo[v];
        tile[q][16 + rbase + v][ccol] = c_hi[v];
    }
    __syncthreads();

    // Fixed-order reduction over the 8 K-chunks, then store 32x32 tile.
    for (int e = tid; e < 32 * 32; e += 512) {
        const int r = e >> 5, c = e & 31;
        float s = 0.f;
        #pragma unroll
        for (int qq = 0; qq < 8; ++qq) s += tile[qq][r][c];
        out[(size_t)(r * N_LAYERS + l) * D_ACTS + n0 + c] = s;
    }
}

// ---------------------------------------------------------------------------
// Kernel 2: dec_norms + tanh partial sums.
// Grid: 12 k-slices x 48 d-tiles of 128. Block: 256 thr = 8 waves.
// ---------------------------------------------------------------------------
__global__ __launch_bounds__(256) void norm_tanh_partial(
    const float* __restrict__ feat, WPtrs wp, float* __restrict__ partial)
{
    __shared__ float dn[128];
    __shared__ float red[8];

    const int k  = blockIdx.x / 48;
    const int d0 = (blockIdx.x % 48) * 128;
    const int tid  = threadIdx.x;
    const int wave = __builtin_amdgcn_readfirstlane(tid >> 5); // SGPR wave id
    const int lane = tid & 31;

    // Phase 1: dec_norms[k, d0+dd] = sum_{l>=k} ||W_l[k, d, :]||
    for (int dd = wave; dd < 128; dd += 8) {
        const int d = d0 + dd;
        float norm = 0.f;
        for (int l = k; l < N_LAYERS; ++l) {
            const float* row = wp.w[l] + (size_t)(k * D_FEAT + d) * D_ACTS;
            float s = 0.f;
            #pragma unroll
            for (int j = 0; j < 6; ++j) {           // 32 lanes * 4 * 6 = 768
                float4 v4 = *(const float4*)(row + lane * 4 + j * 128);
                s += v4.x * v4.x + v4.y * v4.y + v4.z * v4.z + v4.w * v4.w;
            }
            #pragma unroll
            for (int off = 16; off; off >>= 1) s += __shfl_xor(s, off, 32);
            norm += sqrtf(s);
        }
        if (lane == 0) dn[dd] = norm;
    }
    __syncthreads();

    // Phase 2: acc = sum_{b,dd} tanh(F[b,k,d] * dn[dd] * C)
    float acc = 0.f;
    #pragma unroll
    for (int it = 0; it < 16; ++it) {                // 32*128 / 256 threads
        const int idx = it * 256 + tid;
        const int b  = idx >> 7;
        const int dd = idx & 127;
        const float f = feat[(size_t)b * LSTR + k * D_FEAT + d0 + dd];
        acc += tanhf(f * dn[dd] * 0.1f);
    }
    #pragma unroll
    for (int off = 16; off; off >>= 1) acc += __shfl_xor(acc, off, 32);
    if (lane == 0) red[wave] = acc;
    __syncthreads();
    if (tid == 0) {
        float t = 0.f;
        #pragma unroll
        for (int wv = 0; wv < 8; ++wv) t += red[wv];
        partial[blockIdx.x] = t;
    }
}

// ---------------------------------------------------------------------------
// Kernel 3: fixed-order reduction of per-block partials -> sparsity scalar.
// ---------------------------------------------------------------------------
__global__ __launch_bounds__(256) void reduce_sparsity(
    const float* __restrict__ partial, int n, float* __restrict__ out)
{
    __shared__ float buf[256];
    float s = 0.f;
    for (int i = threadIdx.x; i < n; i += 256) s += partial[i];
    buf[threadIdx.x] = s;
    __syncthreads();
    for (int stride = 128; stride; stride >>= 1) {
        if (threadIdx.x < stride) buf[threadIdx.x] += buf[threadIdx.x + stride];
        __syncthreads();
    }
    if (threadIdx.x == 0) out[0] = 0.1f * buf[0] / (float)BATCH;
}

// ---------------------------------------------------------------------------
extern "C" void kernel_launch(void* const* d_in, const int* in_sizes, int n_in,
                              void* d_out, int out_size, void* d_ws, size_t ws_size,
                              hipStream_t stream)
{
    const float* feat = (const float*)d_in[0];
    WPtrs wp;
    for (int i = 0; i < N_LAYERS; ++i) wp.w[i] = (const float*)d_in[1 + i];

    float* recons   = (float*)d_out;                       // 32*12*768
    float* sparsity = recons + BATCH * N_LAYERS * D_ACTS;  // 1 float
    float* partial  = (float*)d_ws;                        // 576 floats

    // GEMM: 12 layers * 24 column tiles
    decoder_gemm<<<dim3(N_LAYERS * 24), dim3(512), 0, stream>>>(feat, wp, recons);

    // Norms + tanh partials: 12 * 48 blocks
    const int nPart = N_LAYERS * 48;
    norm_tanh_partial<<<dim3(nPart), dim3(256), 0, stream>>>(feat, wp, partial);

    // Final deterministic reduction
    reduce_sparsity<<<dim3(1), dim3(256), 0, stream>>>(partial, nPart, sparsity);
}